// EntmaxAlphaActivation_25271587570066
// MI455X (gfx1250) — compile-verified
//
#include <hip/hip_runtime.h>
#include <cstdint>
#include <cstddef>
#include <cmath>

// ---------------------------------------------------------------------------
// entmax_bisect, one workgroup per row of 4096 fp32 scores.
//  - gfx1250 async path: global_load_async_to_lds_b128 + s_wait_asynccnt
//  - row held in VGPRs across all 50 bisection iterations (zero re-reads)
//  - v_wmma_f32_16x16x4_f32 performs the 32->16 lane contraction of each
//    row-sum (A = 0.5 broadcast, B = partials in both VGPRs: each output
//    column = s_j + s_{j+16}, replicated across half-waves; layout-robust
//    since the K-contraction sums all 4 B slots of a column)
//  - remaining 16-lane sum + cross-wave combine via shuffles + LDS broadcast
//  - exact fast path for exponent==2 (alpha==1.5): z*z instead of exp2/log2
// ---------------------------------------------------------------------------

#define ROWLEN   4096
#define THREADS  256
#define EPT      (ROWLEN / THREADS)   // 16 elements per thread
#define NWAVES   (THREADS / 32)       // 8 wave32s per block
#define N_ITER   50
#define ALPHA_MIN 1.001f

#if __has_builtin(__builtin_amdgcn_exp2f) && __has_builtin(__builtin_amdgcn_logf)
#define FAST_EXP2(x) __builtin_amdgcn_exp2f(x)   // v_exp_f32
#define FAST_LOG2(x) __builtin_amdgcn_logf(x)    // v_log_f32 (base-2 on AMD)
#else
#define FAST_EXP2(x) exp2f(x)
#define FAST_LOG2(x) log2f(x)
#endif

typedef __attribute__((ext_vector_type(2))) float v2f;
typedef __attribute__((ext_vector_type(8))) float v8f;

struct __align__(16) Smem {
  float     scores[ROWLEN];        // 16 KB staged row
  uint32_t  mask32[ROWLEN / 4];    // 4 KB staged mask bytes (as dwords)
  float     red[NWAVES];           // cross-wave reduction scratch
};

// ---- wave-level sum: WMMA contraction (32->16) + 4 shuffle steps ----------
__device__ __forceinline__ float wave_sum(float v) {
#if __has_builtin(__builtin_amdgcn_wmma_f32_16x16x4_f32)
  v2f a = {0.5f, 0.5f};            // uniform A: every entry 0.5
  v2f b = {v, v};                  // lane partial in both B VGPRs
  v8f c = {0.f, 0.f, 0.f, 0.f, 0.f, 0.f, 0.f, 0.f};
  // D[i][j] = 0.5 * (s_j + s_j + s_{j+16} + s_{j+16}) = s_j + s_{j+16}
  v8f d = __builtin_amdgcn_wmma_f32_16x16x4_f32(
      /*neg_a=*/false, a, /*neg_b=*/false, b,
      /*c_mod=*/(short)0, c, /*reuse_a=*/false, /*reuse_b=*/false);
  float t = d[0];                  // every lane l holds t_{l&15}
#pragma unroll
  for (int o = 8; o > 0; o >>= 1) t += __shfl_xor(t, o, 32);
  return t;                        // full 32-lane sum, uniform across wave
#else
  float t = v;
#pragma unroll
  for (int o = 16; o > 0; o >>= 1) t += __shfl_xor(t, o, 32);
  return t;
#endif
}

__device__ __forceinline__ float wave_max(float v) {
#pragma unroll
  for (int o = 16; o > 0; o >>= 1) v = fmaxf(v, __shfl_xor(v, o, 32));
  return v;
}

// All-reduce across the 256-thread block; every thread returns the result.
// Fixed combine order => deterministic across calls.
__device__ __forceinline__ float block_sum(float v, float* buf) {
  v = wave_sum(v);
  const int w = threadIdx.x >> 5, l = threadIdx.x & 31;
  __syncthreads();                       // protect buf from previous round's reads
  if (l == 0) buf[w] = v;
  __syncthreads();
  float t = 0.f;
#pragma unroll
  for (int i = 0; i < NWAVES; ++i) t += buf[i];   // LDS same-address broadcast
  return t;
}
__device__ __forceinline__ float block_max(float v, float* buf) {
  v = wave_max(v);
  const int w = threadIdx.x >> 5, l = threadIdx.x & 31;
  __syncthreads();
  if (l == 0) buf[w] = v;
  __syncthreads();
  float t = -INFINITY;
#pragma unroll
  for (int i = 0; i < NWAVES; ++i) t = fmaxf(t, buf[i]);
  return t;
}

template <bool SQ>
__device__ __forceinline__ float pofz(float z, float ex) {
  // z > 0 guaranteed by caller
  if (SQ) return z * z;                      // exponent == 2 (alpha == 1.5)
  return FAST_EXP2(ex * FAST_LOG2(z));
}

template <bool SQ>
__device__ __forceinline__ void entmax_row(const float* __restrict__ srow,
                                           const uint8_t* __restrict__ mrow,
                                           float* __restrict__ orow,
                                           float am1, float ex, Smem& sm) {
  const int t = threadIdx.x;

  // ---- stage scores + mask into LDS via CDNA5 async DMA path --------------
  {
    const uint64_t sbase = (uint64_t)(uintptr_t)srow;
    const uint64_t mbase = (uint64_t)(uintptr_t)mrow;
    const uint32_t lds_s = (uint32_t)(uintptr_t)(&sm.scores[0]);
    const uint32_t lds_m = (uint32_t)(uintptr_t)(&sm.mask32[0]);
#pragma unroll
    for (int k = 0; k < 4; ++k) {          // 4 x 16B chunks of scores / thread
      uint32_t off = (uint32_t)(t + k * THREADS) * 16u;
      asm volatile("global_load_async_to_lds_b128 %0, %1, %2"
                   :: "v"(lds_s + off), "v"(off), "s"(sbase) : "memory");
    }
    {
      uint32_t moff = (uint32_t)t * 16u;   // 1 x 16B chunk of mask / thread
      asm volatile("global_load_async_to_lds_b128 %0, %1, %2"
                   :: "v"(lds_m + moff), "v"(moff), "s"(mbase) : "memory");
    }
    asm volatile("s_wait_asynccnt 0x0" ::: "memory");
  }
  __syncthreads();

  // ---- pull 16 contiguous elements into registers, apply mask -------------
  float Xs[EPT];
  {
    const float4* s4 = (const float4*)(&sm.scores[t * EPT]);
#pragma unroll
    for (int k = 0; k < EPT / 4; ++k) {
      float4   v  = s4[k];
      uint32_t mw = sm.mask32[t * (EPT / 4) + k];   // 4 mask bytes
      Xs[4 * k + 0] = (mw & 0x000000FFu) ? v.x * am1 : -INFINITY;
      Xs[4 * k + 1] = (mw & 0x0000FF00u) ? v.y * am1 : -INFINITY;
      Xs[4 * k + 2] = (mw & 0x00FF0000u) ? v.z * am1 : -INFINITY;
      Xs[4 * k + 3] = (mw & 0xFF000000u) ? v.w * am1 : -INFINITY;
    }
  }

  // ---- row max, bisection bounds ------------------------------------------
  float mx = -INFINITY;
#pragma unroll
  for (int e = 0; e < EPT; ++e) mx = fmaxf(mx, Xs[e]);
  mx = block_max(mx, sm.red);

  float tau_lo = mx - 1.0f;                                   // _gp(1, alpha)
  float tau_hi = mx - powf(1.0f / (float)ROWLEN, am1);        // _gp(1/d, alpha)
  float dm     = tau_hi - tau_lo;

  // f_lo = sum p(tau_lo) - 1  (fixed across iterations, per reference)
  float part = 0.f;
#pragma unroll
  for (int e = 0; e < EPT; ++e) {
    float z = Xs[e] - tau_lo;
    part += (z > 0.f) ? pofz<SQ>(z, ex) : 0.f;
  }
  const float f_lo = block_sum(part, sm.red) - 1.0f;

  // ---- 50 bisection iterations, row resident in VGPRs ---------------------
  float p[EPT];
  float ssum = 0.f;
  for (int it = 0; it < N_ITER; ++it) {
    dm *= 0.5f;
    const float tau_m = tau_lo + dm;
    float acc = 0.f;
#pragma unroll
    for (int e = 0; e < EPT; ++e) {
      float z  = Xs[e] - tau_m;
      float pe = (z > 0.f) ? pofz<SQ>(z, ex) : 0.f;
      p[e] = pe;
      acc += pe;
    }
    ssum = block_sum(acc, sm.red);           // uniform across block
    const float f_m = ssum - 1.0f;
    if (f_m * f_lo >= 0.f) tau_lo = tau_m;   // uniform branch
  }

  // ---- normalize last p_m and store (ensure_sum_one) ----------------------
  const float inv = 1.0f / ssum;
  float4* o4 = (float4*)(&orow[t * EPT]);
#pragma unroll
  for (int k = 0; k < EPT / 4; ++k) {
    float4 v;
    v.x = p[4 * k + 0] * inv;
    v.y = p[4 * k + 1] * inv;
    v.z = p[4 * k + 2] * inv;
    v.w = p[4 * k + 3] * inv;
    o4[k] = v;
  }
}

__global__ __launch_bounds__(THREADS)
void EntmaxAlphaActivation_kernel(const float* __restrict__ scores,
                                  const uint8_t* __restrict__ mask,
                                  const float* __restrict__ alpha,
                                  float* __restrict__ out, int rows) {
  __shared__ Smem sm;
  const int row = blockIdx.x;
  if (row >= rows) return;

  const float a   = alpha[0];                 // uniform scalar
  const float ac  = fmaxf(a, ALPHA_MIN);
  const float am1 = ac - 1.0f;
  const float ex  = 1.0f / am1;

  const float*   srow = scores + (size_t)row * ROWLEN;
  const uint8_t* mrow = mask   + (size_t)row * ROWLEN;
  float*         orow = out    + (size_t)row * ROWLEN;

  if (ex == 2.0f) entmax_row<true >(srow, mrow, orow, am1, ex, sm);
  else            entmax_row<false>(srow, mrow, orow, am1, ex, sm);
}

extern "C" void kernel_launch(void* const* d_in, const int* in_sizes, int n_in,
                              void* d_out, int out_size, void* d_ws, size_t ws_size,
                              hipStream_t stream) {
  (void)n_in; (void)d_ws; (void)ws_size; (void)out_size;
  const float*   scores = (const float*)d_in[0];
  const uint8_t* mask   = (const uint8_t*)d_in[1];   // jax bool -> 1 byte/elem
  const float*   alpha  = (const float*)d_in[2];
  float*         out    = (float*)d_out;

  const int rows = in_sizes[0] / ROWLEN;             // 4096
  EntmaxAlphaActivation_kernel<<<rows, THREADS, 0, stream>>>(scores, mask, alpha,
                                                             out, rows);
}